// LSTM1_6682969112846
// MI455X (gfx1250) — compile-verified
//
#include <hip/hip_runtime.h>
#include <hip/hip_bf16.h>

typedef __attribute__((ext_vector_type(2))) float v2f;
typedef __attribute__((ext_vector_type(8))) float v8f;

// ---------------------------------------------------------------------------
// Phase 1: xg[r, g] = x[r, 0:6] . w_ih[g, 0:6] + (b_ih[g] + b_hh[g])
// r = b*T + t flattened (8.4M rows), g = 0..7.  V_WMMA_F32_16X16X4_F32:
// 16 rows per tile, K=6 padded to 8 -> two chained K=4 WMMAs, N columns 0..7
// hold the 8 gates (8..15 padded with zero weights).
// Loop body is branchless: K-pair selection is done via address arithmetic
// and v_cndmask, not exec masking, so the hot loop is straight-line
// (2x global_load_b64 -> 2x v_wmma -> stores).
// ---------------------------------------------------------------------------
__global__ void lstm_xg_wmma_kernel(const float* __restrict__ x,
                                    const float* __restrict__ w_ih,
                                    const float* __restrict__ b_ih,
                                    const float* __restrict__ b_hh,
                                    float* __restrict__ xg,
                                    long long nrows) {
  const int lane    = threadIdx.x & 31;
  const int n       = lane & 15;        // N column (gate) / M row for A loads
  const bool hiHalf = lane >= 16;       // selects K pair (A) / row half (C)

  const long long waveGlobal =
      (long long)blockIdx.x * (blockDim.x >> 5) + (threadIdx.x >> 5);
  const long long nWaves = (long long)gridDim.x * (blockDim.x >> 5);

  // B matrix = w_ih^T padded to 8x16 (two 4x16 K-tiles).
  // Layout mirrors A: lanes 0-15 hold K0/K1 in v.x/v.y, lanes 16-31 hold K2/K3.
  v2f bm1 = {0.f, 0.f}, bm2 = {0.f, 0.f};
  if (n < 8) {
    const float* wr = w_ih + n * 6;     // gate n, features 0..5
    if (!hiHalf) { bm1.x = wr[0]; bm1.y = wr[1]; bm2.x = wr[4]; bm2.y = wr[5]; }
    else         { bm1.x = wr[2]; bm1.y = wr[3]; /* features 6,7 are zero pad */ }
  }
  const float bias = (n < 8) ? (b_ih[n] + b_hh[n]) : 0.f;

  const int kb = hiHalf ? 2 : 0;        // first feature of this lane's K pair
  const long long ntiles = nrows >> 4;

  for (long long tile = waveGlobal; tile < ntiles; tile += nWaves) {
    const long long r0 = tile << 4;

    // A matrix: row M = lane&15.  Row stride = 24 B -> every row base is
    // 8-byte aligned.  All loads unconditional & in-bounds; pad via select.
    const float* xr = x + (r0 + n) * 6;
    v2f a1;
    a1.x = xr[kb];
    a1.y = xr[kb + 1];
    const float t0 = xr[4];
    const float t1 = xr[5];
    v2f a2;
    a2.x = hiHalf ? 0.f : t0;           // K=2,3 of second tile = features 6,7 = 0
    a2.y = hiHalf ? 0.f : t1;

    v8f c;
#pragma unroll
    for (int j = 0; j < 8; ++j) c[j] = bias;   // bias depends only on column n

    c = __builtin_amdgcn_wmma_f32_16x16x4_f32(false, a1, false, bm1,
                                              (short)0, c, false, false);
    c = __builtin_amdgcn_wmma_f32_16x16x4_f32(false, a2, false, bm2,
                                              (short)0, c, false, false);

    if (n < 8) {
      const long long rowBase = r0 + (hiHalf ? 8 : 0);
#pragma unroll
      for (int j = 0; j < 8; ++j) {
        // C layout: VGPR j holds rows M=j (lanes 0-15) and M=j+8 (lanes 16-31)
        xg[(rowBase + j) * 8 + n] = c[j];
      }
    }
  }
}

// ---------------------------------------------------------------------------
// Phase 2+3: LSTM recurrence + fused FC head.
// 8 lanes per batch element (one per gate: i0,i1,f0,f1,g0,g1,o0,o1),
// 4 batch elements per wave.  Cross-lane combine via wave shuffles.
// ---------------------------------------------------------------------------
__global__ void lstm_rec_fc_kernel(const float* __restrict__ xg,
                                   const float* __restrict__ w_hh,
                                   const float* __restrict__ fc1_w,
                                   const float* __restrict__ fc1_b,
                                   const float* __restrict__ fc_w,
                                   const float* __restrict__ fc_b,
                                   float* __restrict__ out,
                                   int Bn, int Tn) {
  const int lane = threadIdx.x & 31;
  const int g    = lane & 7;        // gate index 0..7
  const int sub  = lane >> 3;       // batch slot within wave 0..3
  const long long wave =
      (long long)blockIdx.x * (blockDim.x >> 5) + (threadIdx.x >> 5);
  const int b = (int)(wave * 4 + sub);
  if (b >= Bn) return;              // exact launch: never divergent

  const float wh0 = w_hh[g * 2 + 0];
  const float wh1 = w_hh[g * 2 + 1];
  const bool isTanh = (g >> 1) == 2;                // gates g0,g1 use tanh
  const float kNegLog2e = isTanh ? -2.8853900817779268f   // -2/ln2
                                 : -1.4426950408889634f;  // -1/ln2

  const float* xp = xg + (long long)b * Tn * 8 + g;
  const int base = lane & ~7;       // first lane of this batch group
  const int j    = lane & 1;        // hidden unit this lane tracks (0/1)

  float h0 = 0.f, h1 = 0.f, cst = 0.f;

  for (int t = 0; t < Tn; ++t) {
    // prefetch ~64 timesteps (2 KiB / batch stream) ahead
    __builtin_prefetch(xp + (long long)(t + 64) * 8, 0, 0);

    const float pre = xp[(long long)t * 8] + wh0 * h0 + wh1 * h1;
    // sigmoid(x) = 1/(1+2^(-x*log2e)); tanh(x) = 2*sigmoid(2x)-1
    const float e   = __builtin_amdgcn_exp2f(pre * kNegLog2e);
    const float s   = __builtin_amdgcn_rcpf(1.f + e);
    const float act = isTanh ? (2.f * s - 1.f) : s;

    const float iv = __shfl(act, base + 0 + j, 32);
    const float fv = __shfl(act, base + 2 + j, 32);
    const float gv = __shfl(act, base + 4 + j, 32);
    const float ov = __shfl(act, base + 6 + j, 32);

    cst = fv * cst + iv * gv;                       // c for hidden unit j
    const float e2 = __builtin_amdgcn_exp2f(cst * -2.8853900817779268f);
    const float th = 2.f * __builtin_amdgcn_rcpf(1.f + e2) - 1.f;
    const float h  = ov * th;

    h0 = __shfl(h, base + 0, 32);                   // rebroadcast hidden state
    h1 = __shfl(h, base + 1, 32);
  }

  // ---- fused FC head: out = relu(relu(hn) @ fc1_w^T + fc1_b) @ fc_w^T + fc_b
  const float r0 = fmaxf(h0, 0.f);
  const float r1 = fmaxf(h1, 0.f);
  float acc = 0.f;
#pragma unroll
  for (int k = 0; k < 16; ++k) {                    // 128 units / 8 lanes
    const int jj = g + 8 * k;
    float v = r0 * fc1_w[jj * 2 + 0] + r1 * fc1_w[jj * 2 + 1] + fc1_b[jj];
    v = fmaxf(v, 0.f);
    acc += v * fc_w[jj];
  }
  acc += __shfl_xor(acc, 1, 32);                    // reduce the 8-lane group
  acc += __shfl_xor(acc, 2, 32);
  acc += __shfl_xor(acc, 4, 32);
  if (g == 0) out[b] = acc + fc_b[0];
}

// ---------------------------------------------------------------------------
extern "C" void kernel_launch(void* const* d_in, const int* in_sizes, int n_in,
                              void* d_out, int out_size, void* d_ws,
                              size_t ws_size, hipStream_t stream) {
  const float* x     = (const float*)d_in[0];
  const float* w_ih  = (const float*)d_in[1];
  const float* w_hh  = (const float*)d_in[2];
  const float* b_ih  = (const float*)d_in[3];
  const float* b_hh  = (const float*)d_in[4];
  const float* fc1_w = (const float*)d_in[5];
  const float* fc1_b = (const float*)d_in[6];
  const float* fc_w  = (const float*)d_in[7];
  const float* fc_b  = (const float*)d_in[8];
  float* out = (float*)d_out;

  const int Bn = out_size;                       // 4096
  const int Tn = in_sizes[0] / (Bn * 6);         // 2048
  const long long nrows = (long long)Bn * Tn;    // 8.4M (b,t) rows

  float* xg = (float*)d_ws;                      // B*T*8 f32 = 256 MiB scratch

  // Phase 1: WMMA input projection. 32768 waves, 16 row-tiles each.
  lstm_xg_wmma_kernel<<<4096, 256, 0, stream>>>(x, w_ih, b_ih, b_hh, xg, nrows);

  // Phase 2+3: recurrence + FC. 1024 waves = Bn/4 exactly.
  const int waves = (Bn + 3) / 4;
  const int blocks = (waves + 7) / 8;            // 8 waves (256 thr) per block
  lstm_rec_fc_kernel<<<blocks, 256, 0, stream>>>(xg, w_hh, fc1_w, fc1_b,
                                                 fc_w, fc_b, out, Bn, Tn);
}